// QuantizedCompressor_1211180777498
// MI455X (gfx1250) — compile-verified
//
#include <hip/hip_runtime.h>
#include <stdint.h>

#define H_DIM   4096
#define H4      1024        // float4 per row
#define GROUPS  32
#define G4      32          // float4 per 128-col group
#define QMAX_F  255.0f
#define EPS_F   1e-5f

typedef float v4f __attribute__((ext_vector_type(4)));

// ---- order-preserving float<->uint map (so we can use integer atomics) ----
__device__ __forceinline__ unsigned f2ord(float f) {
  unsigned u = __float_as_uint(f);
  return (u & 0x80000000u) ? ~u : (u | 0x80000000u);
}
__device__ __forceinline__ float ord2f(unsigned u) {
  u = (u & 0x80000000u) ? (u ^ 0x80000000u) : ~u;
  return __uint_as_float(u);
}

// ---- CDNA5 async global->LDS copy (16B per lane), tracked by ASYNCcnt ----
__device__ __forceinline__ void async_ld16(unsigned ldsOff, const float* g) {
  asm volatile("global_load_async_to_lds_b128 %0, %1, off"
               :: "v"(ldsOff), "v"((unsigned long long)(uintptr_t)g)
               : "memory");
}
__device__ __forceinline__ unsigned lds_off(const void* p) {
  return (unsigned)(uintptr_t)p;   // addrspace(3) offset survives truncation
}

// ---------------- kernel 0: init workspace atomics ----------------
// mm layout (uint): [tensor][0..31]=ordered-min slots, [tensor][32..63]=ordered-max slots
__global__ void init_ws(unsigned* mm) {
  int t = threadIdx.x;
  if (t < 128) mm[t] = ((t & 63) < 32) ? 0xFFFFFFFFu : 0u;
}

// ---------------- kernel 1: per-group min/max over all rows ----------------
// grid: (rowBlocks, 32 groups, 2 tensors), block 256 (=8 wave32)
__global__ void __launch_bounds__(256)
group_minmax(const float* __restrict__ in0, const float* __restrict__ in1,
             unsigned* __restrict__ mm, int nRows, int rowsPerBlock) {
  const int tid  = threadIdx.x;
  const int lane = tid & 31;
  const int wid  = tid >> 5;                 // wave id 0..7
  const int g    = blockIdx.y;
  const int tensor = blockIdx.z;
  const float4* __restrict__ in4 = (const float4*)(tensor ? in1 : in0);
  const int col4 = g * G4 + lane;            // 32 lanes cover one 512B group row

  int row0 = blockIdx.x * rowsPerBlock;
  int row1 = row0 + rowsPerBlock; if (row1 > nRows) row1 = nRows;

  float mn = __int_as_float(0x7F800000);     // +inf
  float mx = __int_as_float(0xFF800000);     // -inf
#pragma unroll 4
  for (int r = row0 + wid; r < row1; r += 8) {
    float4 v = in4[(size_t)r * H4 + col4];
    mn = fminf(mn, fminf(fminf(v.x, v.y), fminf(v.z, v.w)));
    mx = fmaxf(mx, fmaxf(fmaxf(v.x, v.y), fmaxf(v.z, v.w)));
  }
  // wave32 butterfly reduce
  for (int off = 16; off > 0; off >>= 1) {
    mn = fminf(mn, __shfl_xor(mn, off, 32));
    mx = fmaxf(mx, __shfl_xor(mx, off, 32));
  }
  __shared__ float smn[8], smx[8];
  if (lane == 0) { smn[wid] = mn; smx[wid] = mx; }
  __syncthreads();
  if (tid == 0) {
    for (int w = 1; w < 8; ++w) { mn = fminf(mn, smn[w]); mx = fmaxf(mx, smx[w]); }
    atomicMin(&mm[tensor * 64 + g],      f2ord(mn));
    atomicMax(&mm[tensor * 64 + 32 + g], f2ord(mx));
  }
}

// ---------------- kernel 2: scales / zero-points / reciprocals ----------------
// fs layout (float): per tensor stride 96: [g]=scale, [32+g]=zp, [64+g]=1/scale
__global__ void compute_scales(const unsigned* __restrict__ mm, float* __restrict__ fs) {
  int t = threadIdx.x;
  if (t >= 64) return;
  int tensor = t >> 5, g = t & 31;
  float mn = ord2f(mm[tensor * 64 + g]);
  float mx = ord2f(mm[tensor * 64 + 32 + g]);
  if (mx == mn) mx = mn + EPS_F;
  float s  = (mx - mn) / QMAX_F;
  float zp = -rintf(mn / s);                 // QMIN == 0
  fs[tensor * 96 + g]      = s;
  fs[tensor * 96 + 32 + g] = zp;
  fs[tensor * 96 + 64 + g] = 1.0f / s;
}

// ---------------- kernel 3: quantize-dequantize, async-LDS double buffered ----
// Each block streams chunks of 512 float4 (8 KB); each thread handles 2 float4/chunk.
__global__ void __launch_bounds__(256)
qdq_kernel(const float* __restrict__ in, float* __restrict__ out,
           const float* __restrict__ stab, size_t n4, int chunks) {
  __shared__ v4f  buf[2][512];               // 16 KB double buffer
  __shared__ float tabS[32], tabZ[32], tabI[32];
  const int tid = threadIdx.x;
  if (tid < 32) {                            // stage scale table in LDS once
    tabS[tid] = stab[tid];
    tabZ[tid] = stab[32 + tid];
    tabI[tid] = stab[64 + tid];
  }
  __syncthreads();

  const size_t base4 = (size_t)blockIdx.x * (size_t)chunks * 512;

  if (base4 + (size_t)chunks * 512 <= n4) {
    // ---- fully in-range: async double-buffered pipeline ----
    async_ld16(lds_off(&buf[0][tid]),       in + (base4 + tid) * 4);
    async_ld16(lds_off(&buf[0][tid + 256]), in + (base4 + 256 + tid) * 4);
    for (int c = 0; c < chunks; ++c) {
      const int cur = c & 1;
      if (c + 1 < chunks) {
        size_t j = base4 + (size_t)(c + 1) * 512 + tid;
        async_ld16(lds_off(&buf[cur ^ 1][tid]),       in + j * 4);
        async_ld16(lds_off(&buf[cur ^ 1][tid + 256]), in + (j + 256) * 4);
        asm volatile("s_wait_asynccnt 2" ::: "memory");  // chunk c resident
      } else {
        asm volatile("s_wait_asynccnt 0" ::: "memory");
      }
      size_t i0 = base4 + (size_t)c * 512 + tid;
      size_t i1 = i0 + 256;
#pragma unroll
      for (int e = 0; e < 2; ++e) {
        size_t i = e ? i1 : i0;
        v4f v = buf[cur][tid + e * 256];     // ds_load_b128, own slot
        int g = (int)((i >> 5) & (GROUPS - 1));
        float s  = tabS[g];
        float zp = tabZ[g];
        float is = tabI[g];
        v4f r;
        r.x = (fminf(fmaxf(rintf(__fmaf_rn(v.x, is, zp)), 0.0f), QMAX_F) - zp) * s;
        r.y = (fminf(fmaxf(rintf(__fmaf_rn(v.y, is, zp)), 0.0f), QMAX_F) - zp) * s;
        r.z = (fminf(fmaxf(rintf(__fmaf_rn(v.z, is, zp)), 0.0f), QMAX_F) - zp) * s;
        r.w = (fminf(fmaxf(rintf(__fmaf_rn(v.w, is, zp)), 0.0f), QMAX_F) - zp) * s;
        __builtin_nontemporal_store(r, (v4f*)(out + i * 4));  // TH=NT write-stream
      }
    }
  } else {
    // ---- tail block (not hit for these shapes): simple guarded path ----
    for (int c = 0; c < 2 * chunks; ++c) {
      size_t i = base4 + (size_t)c * 256 + tid;
      if (i >= n4) break;
      v4f v = *(const v4f*)(in + i * 4);
      int g = (int)((i >> 5) & (GROUPS - 1));
      float s = tabS[g], zp = tabZ[g], is = tabI[g];
      v4f r;
      r.x = (fminf(fmaxf(rintf(__fmaf_rn(v.x, is, zp)), 0.0f), QMAX_F) - zp) * s;
      r.y = (fminf(fmaxf(rintf(__fmaf_rn(v.y, is, zp)), 0.0f), QMAX_F) - zp) * s;
      r.z = (fminf(fmaxf(rintf(__fmaf_rn(v.z, is, zp)), 0.0f), QMAX_F) - zp) * s;
      r.w = (fminf(fmaxf(rintf(__fmaf_rn(v.w, is, zp)), 0.0f), QMAX_F) - zp) * s;
      __builtin_nontemporal_store(r, (v4f*)(out + i * 4));
    }
  }
}

extern "C" void kernel_launch(void* const* d_in, const int* in_sizes, int n_in,
                              void* d_out, int out_size, void* d_ws, size_t ws_size,
                              hipStream_t stream) {
  const float* keys = (const float*)d_in[0];
  const float* vals = (const float*)d_in[1];
  float* out = (float*)d_out;
  const size_t n = (size_t)in_sizes[0];      // 4*4096*4096 = 67,108,864 per tensor
  const int nRows = (int)(n / H_DIM);        // 16384

  unsigned* mm = (unsigned*)d_ws;            // 128 uints (512 B)
  float* fs = (float*)((char*)d_ws + 512);   // 192 floats (768 B)

  init_ws<<<1, 128, 0, stream>>>(mm);

  const int ROW_BLOCKS = 64;
  int rowsPerBlock = (nRows + ROW_BLOCKS - 1) / ROW_BLOCKS;
  dim3 gridR(ROW_BLOCKS, GROUPS, 2);
  group_minmax<<<gridR, 256, 0, stream>>>(keys, vals, mm, nRows, rowsPerBlock);

  compute_scales<<<1, 64, 0, stream>>>(mm, fs);

  const size_t n4 = n / 4;                   // 16,777,216 float4 per tensor
  const int CH = 16;                         // 512 float4 per chunk -> 8 KB/chunk
  int blocks = (int)((n4 + (size_t)CH * 512 - 1) / ((size_t)CH * 512));  // 2048
  qdq_kernel<<<blocks, 256, 0, stream>>>(keys, out,     fs + 0,  n4, CH);
  qdq_kernel<<<blocks, 256, 0, stream>>>(vals, out + n, fs + 96, n4, CH);
}